// ts_skew_14362370638090
// MI455X (gfx1250) — compile-verified
//
#include <hip/hip_runtime.h>
#include <hip/hip_bf16.h>

#define B_DIM 128
#define T_DIM 4096
#define F_DIM 64
#define WINDOW 32
#define S_DIM 4065                         // number of window starts
#define T_CHUNK 192                        // starts handled per block
#define TILE_ROWS (T_CHUNK + WINDOW - 1)   // 223 time rows staged in LDS (57 KB)
#define SUBQ 4
#define PER_THREAD (T_CHUNK / SUBQ)        // 48 outputs per thread

typedef unsigned int u32x4 __attribute__((ext_vector_type(4)));
typedef int i32x8 __attribute__((ext_vector_type(8)));
typedef int i32x4 __attribute__((ext_vector_type(4)));

__global__ __launch_bounds__(256) void ts_skew_kernel(const float* __restrict__ in,
                                                      float* __restrict__ out) {
  __shared__ float tile[TILE_ROWS][F_DIM];   // [time][feature], matches memory order

  const int b  = blockIdx.y;
  const int s0 = blockIdx.x * T_CHUNK;

  // ---------------- TDM async tile load: global -> LDS ----------------
  // Wave 0 issues one tensor DMA for the whole [TILE_ROWS x 64] f32 tile.
  // EXEC is ignored by TENSOR_LOAD_TO_LDS; descriptor values are uniform.
  if (threadIdx.x < 32) {
    unsigned lds_addr = (unsigned)(size_t)(void*)&tile[0][0]; // low 32b of LDS aperture addr
    unsigned long long ga =
        (unsigned long long)(size_t)(const void*)(in + ((size_t)b * T_DIM + (size_t)s0) * F_DIM);

    // D# group 0: count=1 | lds_addr | global_addr[56:0] | type=2
    u32x4 g0;
    g0[0] = 1u;
    g0[1] = lds_addr;
    g0[2] = (unsigned)(ga & 0xffffffffu);
    g0[3] = (unsigned)((ga >> 32) & 0x01ffffffu) | (2u << 30);

    // D# group 1: wg_mask=0, data_size=2 (4B), tensor 64 x (T-s0), tile 64 x 223,
    // dim0 stride = 64 elements. Rows past tensor_dim1 zero-fill (OOB reads = 0),
    // which only feeds guarded (s >= S_DIM) outputs.
    unsigned td0 = F_DIM;
    unsigned td1 = (unsigned)(T_DIM - s0);
    unsigned tl0 = F_DIM;
    unsigned tl1 = TILE_ROWS;
    i32x8 g1;
    g1[0] = (int)(2u << 16);                                  // data_size = 4 bytes
    g1[1] = (int)((td0 & 0xffffu) << 16);                     // tensor_dim0[15:0]
    g1[2] = (int)((td0 >> 16) | ((td1 & 0xffffu) << 16));     // dim0 hi | dim1 lo
    g1[3] = (int)((td1 >> 16) | (tl0 << 16));                 // dim1 hi | tile_dim0
    g1[4] = (int)(tl1 & 0xffffu);                             // tile_dim1 (tile_dim2=0)
    g1[5] = (int)F_DIM;                                       // tensor_dim0_stride[31:0]
    g1[6] = 0;
    g1[7] = 0;
    i32x4 g2 = {0, 0, 0, 0};
    i32x4 g3 = {0, 0, 0, 0};
    i32x8 g4 = {0, 0, 0, 0, 0, 0, 0, 0};   // unused group (<=2D tensor), zero-filled

    __builtin_amdgcn_tensor_load_to_lds(g0, g1, g2, g3, g4, 0);
    __builtin_amdgcn_s_wait_tensorcnt(0);
  }
  __syncthreads();

  // ---------------- rolling-moment skew, one (f, start-range) per thread ----------------
  const int f   = threadIdx.x & (F_DIM - 1);     // lanes 0..31 -> f 0..31: bank-conflict-free
  const int q   = threadIdx.x >> 6;              // 0..3
  const int sl0 = q * PER_THREAD;

  float s1 = 0.f, s2 = 0.f, s3 = 0.f;
#pragma unroll
  for (int k = 0; k < WINDOW; ++k) {
    float x  = tile[sl0 + k][f];
    float x2 = x * x;
    s1 += x; s2 += x2; s3 += x2 * x;
  }

  const int   sbase = s0 + sl0;
  float*      op    = out + ((size_t)b * F_DIM + (size_t)f) * S_DIM + sbase;
  const float inv_w = 1.0f / WINDOW;

  for (int i = 0; i < PER_THREAD; ++i) {
    if (sbase + i < S_DIM) {
      float m1    = s1 * inv_w;
      float m2    = s2 * inv_w;
      float m3    = s3 * inv_w;
      float var   = m2 - m1 * m1 + 1e-6f;
      float third = m3 - 3.0f * m1 * m2 + 2.0f * m1 * m1 * m1;
      float skew  = third / (var * sqrtf(var));
      skew = (skew != skew) ? 0.0f : skew;       // NaN -> 0, per reference
      __builtin_nontemporal_store(skew, op + i); // streamed output, never re-read
    }
    if (i < PER_THREAD - 1) {                    // slide window by one
      float xo = tile[sl0 + i][f];
      float xn = tile[sl0 + i + WINDOW][f];      // max row = 222 < TILE_ROWS
      float xo2 = xo * xo, xn2 = xn * xn;
      s1 += xn - xo;
      s2 += xn2 - xo2;
      s3 += xn2 * xn - xo2 * xo;
    }
  }
}

extern "C" void kernel_launch(void* const* d_in, const int* in_sizes, int n_in,
                              void* d_out, int out_size, void* d_ws, size_t ws_size,
                              hipStream_t stream) {
  const float* in  = (const float*)d_in[0];
  float*       out = (float*)d_out;
  dim3 grid((S_DIM + T_CHUNK - 1) / T_CHUNK, B_DIM);   // 22 x 128 workgroups
  ts_skew_kernel<<<grid, dim3(256), 0, stream>>>(in, out);
}